// ScalarSoftmaxQuantization_36687610642751
// MI455X (gfx1250) — compile-verified
//
#include <hip/hip_runtime.h>

typedef float v2f __attribute__((ext_vector_type(2)));
typedef float v4f __attribute__((ext_vector_type(4)));
typedef float v8f __attribute__((ext_vector_type(8)));

namespace {
constexpr int KK              = 32;          // codebook size
constexpr int NROWS           = 4096 * 512;  // B*L
constexpr int ROWS_PER_WAVE   = 16;          // one WMMA M-tile
constexpr int WAVES_PER_BLOCK = 8;           // 256 threads (wave32)
constexpr int ROWS_PER_BLOCK  = ROWS_PER_WAVE * WAVES_PER_BLOCK; // 128
constexpr int TILE_STRIDE     = 40;          // padded LDS row stride (floats), 160B = 16B-aligned rows
constexpr float kAlpha = -300.0f;
constexpr float kLog2e = 1.44269504088896340736f;
}

__device__ __forceinline__ float fast_exp2(float a) {
#if __has_builtin(__builtin_amdgcn_exp2f)
  return __builtin_amdgcn_exp2f(a);      // v_exp_f32
#else
  return exp2f(a);
#endif
}

__device__ __forceinline__ float fast_rcp(float a) {
#if __has_builtin(__builtin_amdgcn_rcpf)
  return __builtin_amdgcn_rcpf(a);       // v_rcp_f32
#else
  return 1.0f / a;
#endif
}

__global__ __launch_bounds__(256) void ssq_wmma_kernel(
    const float* __restrict__ x, const float* __restrict__ bins,
    float* __restrict__ soft, float* __restrict__ code) {
  // per-wave scratch: D-fragment spill (32 lanes x 8 VGPRs) + padded 16x32 transpose tile
  __shared__ __attribute__((aligned(16))) float lds_c[WAVES_PER_BLOCK][256];
  __shared__ __attribute__((aligned(16))) float lds_t[WAVES_PER_BLOCK][16 * TILE_STRIDE];

  const int tid  = threadIdx.x;
  const int wave = tid >> 5;
  const int lane = tid & 31;
  const int m    = lane & 15;   // WMMA row (M) handled by this lane
  const int h    = lane >> 4;   // K-half selector (A-fragment layout)

  const int rowbase = blockIdx.x * ROWS_PER_BLOCK + wave * ROWS_PER_WAVE;
  const int row     = rowbase + m;   // grid sized exactly: no bounds check needed

  // Lane-private bins in WMMA-A k-order: k = 4*i + 2*h + j  (j = 0,1)
  float bk0[8], bk1[8];
#pragma unroll
  for (int i = 0; i < 8; ++i) {
    v2f b = *(const v2f*)(bins + 4 * i + 2 * h);
    bk0[i] = b.x; bk1[i] = b.y;
  }

  const float xv = x[row];

  // distances + softmax max-subtraction (ALPHA<0 -> subtract min dist)
  float d0[8], d1[8];
  float pmin = 3.402823466e38f;
#pragma unroll
  for (int i = 0; i < 8; ++i) {
    d0[i] = fabsf(xv - bk0[i]);
    d1[i] = fabsf(xv - bk1[i]);
    pmin  = fminf(pmin, fminf(d0[i], d1[i]));
  }
  const float dmin = fminf(pmin, __shfl_xor(pmin, 16));

  const float c2 = kAlpha * kLog2e;   // fold ALPHA and log2(e) into exp2 arg
  float e0[8], e1[8];
#pragma unroll
  for (int i = 0; i < 8; ++i) {
    e0[i] = fast_exp2(c2 * (d0[i] - dmin));
    e1[i] = fast_exp2(c2 * (d1[i] - dmin));
  }

  // Fused reductions via chained V_WMMA_F32_16X16X4_F32:
  //   D[m,0] = sum_k e[m,k]        (B column 0 = ones)
  //   D[m,1] = sum_k e[m,k]*bin_k  (B column 1 = bins, same k-order as A)
  v8f acc = {};
#pragma unroll
  for (int i = 0; i < 8; ++i) {
    v2f a; a.x = e0[i]; a.y = e1[i];
    v2f b;
    b.x = (m == 0) ? 1.0f : ((m == 1) ? bk0[i] : 0.0f);
    b.y = (m == 0) ? 1.0f : ((m == 1) ? bk1[i] : 0.0f);
    acc = __builtin_amdgcn_wmma_f32_16x16x4_f32(
        /*neg_a=*/false, a, /*neg_b=*/false, b,
        /*c_mod=*/(short)0, acc, /*reuse_a=*/false, /*reuse_b=*/false);
  }

  // D fragment holds sums on lanes {0,16}, dots on lanes {1,17}; spill & refetch per row
#pragma unroll
  for (int r = 0; r < 8; ++r) lds_c[wave][lane * 8 + r] = acc[r];
  __syncthreads();
  const int  srcl = (m < 8) ? 0 : 16;
  const float sum = lds_c[wave][srcl * 8 + (m & 7)];
  const float dot = lds_c[wave][(srcl + 1) * 8 + (m & 7)];
  const float rs  = fast_rcp(sum);

  // normalized soft-assignments -> padded LDS tile (transpose for coalesced stores)
#pragma unroll
  for (int i = 0; i < 8; ++i) {
    v2f s; s.x = e0[i] * rs; s.y = e1[i] * rs;
    *(v2f*)&lds_t[wave][m * TILE_STRIDE + 4 * i + 2 * h] = s;
  }
  if (h == 0) __builtin_nontemporal_store(dot * rs, code + row);
  __syncthreads();

  // 4 full-cacheline NT b128 stores per lane: wave writes its 2KB tile contiguously
  float* gt = soft + (size_t)rowbase * KK;
#pragma unroll
  for (int p = 0; p < 4; ++p) {
    const int f = p * 128 + lane * 4;            // float index within compact tile
    const int r = f >> 5, c = f & 31;
    v4f v = *(const v4f*)&lds_t[wave][r * TILE_STRIDE + c];
    __builtin_nontemporal_store(v, (v4f*)(gt + f));
  }
}

extern "C" void kernel_launch(void* const* d_in, const int* in_sizes, int n_in,
                              void* d_out, int out_size, void* d_ws, size_t ws_size,
                              hipStream_t stream) {
  const float* x    = (const float*)d_in[0];
  const float* bins = (const float*)d_in[1];
  float* soft = (float*)d_out;                                // (B,L,K) flat
  float* code = (float*)d_out + (size_t)NROWS * KK;           // (B,L,1) flat
  ssq_wmma_kernel<<<NROWS / ROWS_PER_BLOCK, WAVES_PER_BLOCK * 32, 0, stream>>>(
      x, bins, soft, code);
}